// GATHAConv_6975026888986
// MI455X (gfx1250) — compile-verified
//
#include <hip/hip_runtime.h>
#include <math.h>

#define NND 50000
#define NED 800000
#define HH  3
#define FF  64
#define GN  192            // H*F
#define GK  256
#define NHF (NND*HH*FF)    // 9,600,000
#define NH  (NND*HH)       // 150,000
#define EH  (NED*HH)       // 2,400,000
#define NEG_SLOPE 0.2f

typedef __attribute__((ext_vector_type(2))) float v2f;
typedef __attribute__((ext_vector_type(8))) float v8f;

__device__ __forceinline__ float leaky(float x) {
    return x >= 0.0f ? x : NEG_SLOPE * x;
}

__device__ __forceinline__ void atomicMaxF(float* addr, float val) {
    int* ia = (int*)addr;
    int cur = __float_as_int(*addr);
    while (val > __int_as_float(cur)) {
        int prev = atomicCAS(ia, cur, __float_as_int(val));
        if (prev == cur) break;
        cur = prev;
    }
}

// ---------------- fill ----------------
__global__ void fill_kernel(float* __restrict__ p, float v, size_t n) {
    size_t i = (size_t)blockIdx.x * blockDim.x + threadIdx.x;
    size_t stride = (size_t)gridDim.x * blockDim.x;
    for (; i < n; i += stride) p[i] = v;
}

// ---------------- GEMM: ft = feat (50000x256) @ fc_w (256x192), fp32 WMMA ----------------
// One wave per 16-row M tile; it computes all 12 N tiles (16 cols each) so that
// feat is streamed exactly once. fc_w (192KB) stays cache-resident.
__global__ __launch_bounds__(256) void gemm_wmma_kernel(const float* __restrict__ A,
                                                        const float* __restrict__ B,
                                                        float* __restrict__ C) {
    const int NT = GN / 16;                    // 12 N tiles
    int wave = (int)((blockIdx.x * blockDim.x + threadIdx.x) >> 5);
    if (wave >= NND / 16) return;              // 3125 M tiles, wave-uniform exit
    int lane = threadIdx.x & 31;
    int half = lane >> 4;                      // 0: K pair {0,1}, 1: K pair {2,3}
    int l16  = lane & 15;
    int m0   = wave * 16;

    // A fragment source: row m = m0+l16, columns k0 + 2*half + {0,1}
    const float* arow = A + (size_t)(m0 + l16) * GK + 2 * half;
    // B fragment source: row k0 + 2*half (+1), col = nt*16 + l16
    const float* bbase = B + (size_t)(2 * half) * GN + l16;

    v8f acc[NT];
#pragma unroll
    for (int t = 0; t < NT; ++t) acc[t] = (v8f){0.f,0.f,0.f,0.f,0.f,0.f,0.f,0.f};

    for (int k0 = 0; k0 < GK; k0 += 4) {
        v2f a = *(const v2f*)(arow + k0);      // feat[m][k0+2h], feat[m][k0+2h+1]
#pragma unroll
        for (int t = 0; t < NT; ++t) {
            v2f b;
            b.x = bbase[(size_t)k0 * GN + t * 16];
            b.y = bbase[(size_t)k0 * GN + GN + t * 16];
            acc[t] = __builtin_amdgcn_wmma_f32_16x16x4_f32(
                false, a, false, b, (short)0, acc[t], false, false);
        }
    }

#pragma unroll
    for (int t = 0; t < NT; ++t) {
#pragma unroll
        for (int r = 0; r < 8; ++r) {
            int row = m0 + r + 8 * half;       // C/D layout: VGPR r -> M=r (lanes 0-15), M=r+8 (16-31)
            C[(size_t)row * GN + t * 16 + l16] = acc[t][r];
        }
    }
}

// ---------------- el/er = sum_f ft*attn_{l,r} ----------------
__global__ void node_proj_kernel(const float* __restrict__ ft,
                                 const float* __restrict__ attn_l,
                                 const float* __restrict__ attn_r,
                                 float* __restrict__ el, float* __restrict__ er) {
    int idx = blockIdx.x * blockDim.x + threadIdx.x;
    if (idx >= NH) return;
    int n = idx / HH, h = idx - n * HH;
    size_t base = (size_t)n * GN + h * FF;
    float sl = 0.f, sr = 0.f;
#pragma unroll 4
    for (int f = 0; f < FF; ++f) {
        float v = ft[base + f];
        sl += v * attn_l[h * FF + f];
        sr += v * attn_r[h * FF + f];
    }
    el[idx] = sl; er[idx] = sr;
}

// ---------------- edge pass 1: score + segment max ----------------
__global__ void edge_score_kernel(const float* __restrict__ el, const float* __restrict__ er,
                                  const int* __restrict__ src, const int* __restrict__ dst,
                                  float* __restrict__ ea, float* __restrict__ emax) {
    int idx = blockIdx.x * blockDim.x + threadIdx.x;
    if (idx >= EH) return;
    int e = idx / HH, h = idx - e * HH;
    int s = src[e], d = dst[e];
    float x = leaky(el[s * HH + h] + er[d * HH + h]);
    ea[idx] = x;
    atomicMaxF(&emax[d * HH + h], x);
}

// ---------------- edge pass 2: exp + segment sum ----------------
__global__ void edge_exp_kernel(const int* __restrict__ dst,
                                float* __restrict__ ea,
                                const float* __restrict__ emax,
                                float* __restrict__ esum) {
    int idx = blockIdx.x * blockDim.x + threadIdx.x;
    if (idx >= EH) return;
    int e = idx / HH, h = idx - e * HH;
    int d = dst[e];
    float m = emax[d * HH + h];
    if (!isfinite(m)) m = 0.0f;
    float ee = expf(ea[idx] - m);
    ea[idx] = ee;
    atomicAdd(&esum[d * HH + h], ee);
}

// ---------------- edge pass 3: normalize + gcn_norm_adjust ----------------
__global__ void edge_norm_kernel(const int* __restrict__ dst,
                                 float* __restrict__ ea,
                                 const float* __restrict__ esum,
                                 const float* __restrict__ gcn) {
    int idx = blockIdx.x * blockDim.x + threadIdx.x;
    if (idx >= EH) return;
    int e = idx / HH, h = idx - e * HH;
    int d = dst[e];
    ea[idx] = ea[idx] / fmaxf(esum[d * HH + h], 1e-9f) * gcn[e];
}

// ---------------- aggregation step: hout[dst] += hin[src] * a ----------------
__global__ void agg_kernel(const float* __restrict__ hin, float* __restrict__ hout,
                           const float* __restrict__ ea,
                           const int* __restrict__ src, const int* __restrict__ dst) {
    int idx = blockIdx.x * blockDim.x + threadIdx.x;
    if (idx >= EH) return;
    int e = idx / HH, h = idx - e * HH;
    float w = ea[idx];
    int s = src[e], d = dst[e];
    const float4* pin = (const float4*)(hin + (size_t)s * GN + h * FF);
    float* pout = hout + (size_t)d * GN + h * FF;
#pragma unroll
    for (int f4 = 0; f4 < FF / 4; ++f4) {
        float4 v = pin[f4];
        atomicAdd(pout + 4 * f4 + 0, v.x * w);
        atomicAdd(pout + 4 * f4 + 1, v.y * w);
        atomicAdd(pout + 4 * f4 + 2, v.z * w);
        atomicAdd(pout + 4 * f4 + 3, v.w * w);
    }
}

// ---------------- hop-level attention + output ----------------
__global__ void hop_attn_kernel(const float* __restrict__ ft, const float* __restrict__ h1,
                                const float* __restrict__ h2, const float* __restrict__ h3,
                                const float* __restrict__ pe,
                                const float* __restrict__ hal, const float* __restrict__ har,
                                float* __restrict__ out) {
    int idx = blockIdx.x * blockDim.x + threadIdx.x;
    if (idx >= NH) return;
    int n = idx / HH, h = idx - n * HH;
    const float* hs[4] = { ft, h1, h2, h3 };
    size_t base = (size_t)n * GN + h * FF;
    int pb = h * FF;

    float al = 0.f, ar0 = 0.f, ar1 = 0.f, ar2 = 0.f, ar3 = 0.f;
#pragma unroll 4
    for (int f = 0; f < FF; ++f) {
        float wl = hal[pb + f], wr = har[pb + f];
        float v0 = hs[0][base + f] + pe[0 * GN + pb + f];
        float v1 = hs[1][base + f] + pe[1 * GN + pb + f];
        float v2 = hs[2][base + f] + pe[2 * GN + pb + f];
        float v3 = hs[3][base + f] + pe[3 * GN + pb + f];
        al  += v0 * wl;
        ar0 += v0 * wr; ar1 += v1 * wr; ar2 += v2 * wr; ar3 += v3 * wr;
    }
    float sc[4] = { leaky(ar0 + al), leaky(ar1 + al), leaky(ar2 + al), leaky(ar3 + al) };
    float mx = fmaxf(fmaxf(sc[0], sc[1]), fmaxf(sc[2], sc[3]));
    float ssum = 0.f;
#pragma unroll
    for (int k = 0; k < 4; ++k) { sc[k] = expf(sc[k] - mx); ssum += sc[k]; }
    float inv = 1.0f / ssum;
#pragma unroll
    for (int k = 0; k < 4; ++k) sc[k] *= inv;

#pragma unroll 4
    for (int f = 0; f < FF; ++f) {
        float o = 0.f;
#pragma unroll
        for (int k = 0; k < 4; ++k)
            o += (hs[k][base + f] + pe[k * GN + pb + f]) * sc[k];
        out[base + f] = o;
    }
}

extern "C" void kernel_launch(void* const* d_in, const int* in_sizes, int n_in,
                              void* d_out, int out_size, void* d_ws, size_t ws_size,
                              hipStream_t stream) {
    const float* feat   = (const float*)d_in[0];
    const float* fc_w   = (const float*)d_in[1];
    const float* attn_l = (const float*)d_in[2];
    const float* attn_r = (const float*)d_in[3];
    const float* pe     = (const float*)d_in[4];
    const float* hal    = (const float*)d_in[5];
    const float* har    = (const float*)d_in[6];
    const float* gcn    = (const float*)d_in[7];
    const int*   src    = (const int*)d_in[8];
    const int*   dst    = (const int*)d_in[9];
    float* out = (float*)d_out;

    // workspace layout (floats)
    float* ws   = (float*)d_ws;
    float* ft   = ws;              // NHF
    float* h1   = ft  + NHF;       // NHF
    float* h2   = h1  + NHF;       // NHF
    float* h3   = h2  + NHF;       // NHF  (h1..h3 contiguous)
    float* el   = h3  + NHF;       // NH
    float* er   = el  + NH;        // NH
    float* emax = er  + NH;        // NH
    float* esum = emax+ NH;        // NH
    float* ea   = esum+ NH;        // EH

    const int B = 256;

    // init: h1..h3 = 0 (contiguous), emax = -inf, esum = 0
    fill_kernel<<<2048, B, 0, stream>>>(h1, 0.0f, (size_t)3 * NHF);
    fill_kernel<<<512,  B, 0, stream>>>(emax, -INFINITY, (size_t)NH);
    fill_kernel<<<512,  B, 0, stream>>>(esum, 0.0f, (size_t)NH);

    // 1) ft = feat @ fc_w   (fp32 WMMA)
    {
        int waves = NND / 16;                 // 3125
        int blocks = (waves * 32 + B - 1) / B;
        gemm_wmma_kernel<<<blocks, B, 0, stream>>>(feat, fc_w, ft);
    }

    // 2) el/er
    node_proj_kernel<<<(NH + B - 1) / B, B, 0, stream>>>(ft, attn_l, attn_r, el, er);

    // 3) edge softmax (3 passes, atomics)
    int eb = (EH + B - 1) / B;
    edge_score_kernel<<<eb, B, 0, stream>>>(el, er, src, dst, ea, emax);
    edge_exp_kernel  <<<eb, B, 0, stream>>>(dst, ea, emax, esum);
    edge_norm_kernel <<<eb, B, 0, stream>>>(dst, ea, esum, gcn);

    // 4) K=3 aggregation steps
    agg_kernel<<<eb, B, 0, stream>>>(ft, h1, ea, src, dst);
    agg_kernel<<<eb, B, 0, stream>>>(h1, h2, ea, src, dst);
    agg_kernel<<<eb, B, 0, stream>>>(h2, h3, ea, src, dst);

    // 5) hop attention + output
    hop_attn_kernel<<<(NH + B - 1) / B, B, 0, stream>>>(ft, h1, h2, h3, pe, hal, har, out);
}